// InteractionArch_790273983046
// MI455X (gfx1250) — compile-verified
//
#include <hip/hip_runtime.h>
#include <stdint.h>

// DLRM interaction arch on MI455X (gfx1250):
//   B=65536 samples, F=27 features, D=128 dims.
//   out[b, :] = triu(k=1) of (combined @ combined^T), combined = [dense; sparse].
//
// One wave32 per sample. Stage the 27x128 f32 tile into LDS using CDNA5
// async global->LDS loads (GLOBAL_LOAD_ASYNC_TO_LDS_B128, ASYNCcnt) -- no VGPR
// staging. Rows padded to 32, row stride 132 floats (132 mod 64 = 4) so the
// ds_load_b64 fragment reads are bank-conflict free. Accumulate the three
// unique 16x16 Gram tiles with V_WMMA_F32_16X16X4_F32; the same LDS fragment
// serves as both A (rows) and B (columns) because the WMMA A/B lane layouts
// mirror each other (M <-> N) and the Gram matrix is C * C^T.

typedef float v2f __attribute__((ext_vector_type(2)));
typedef float v8f __attribute__((ext_vector_type(8)));
typedef int   v4i __attribute__((ext_vector_type(4)));

typedef __attribute__((address_space(1))) v4i* gv4i_t;   // global (printed __device__)
typedef __attribute__((address_space(3))) v4i* lv4i_t;   // LDS (printed __shared__)

namespace {
constexpr int kF       = 27;    // features (1 dense + 26 sparse)
constexpr int kD       = 128;   // embedding dim
constexpr int kPairs   = kF * (kF - 1) / 2;  // 351
constexpr int kRows    = 32;    // padded feature rows (two 16-row WMMA tiles)
constexpr int kRStride = 132;   // LDS row stride in floats
constexpr int kWaves   = 4;     // waves (samples) per block
}

// 16B async copy: global -> LDS, tracked by ASYNCcnt, no VGPR staging.
__device__ __forceinline__ void async_copy_b128(const void* gsrc, unsigned lds_byte_off) {
#if __has_builtin(__builtin_amdgcn_global_load_async_to_lds_b128)
    __builtin_amdgcn_global_load_async_to_lds_b128(
        (gv4i_t)(uintptr_t)gsrc,
        (lv4i_t)(uintptr_t)lds_byte_off,
        /*offset=*/0, /*cpol=*/0);
#else
    asm volatile("global_load_async_to_lds_b128 %0, %1, off"
                 :: "v"(lds_byte_off), "v"(gsrc)
                 : "memory");
#endif
}

__device__ __forceinline__ void wait_async_zero() {
#if __has_builtin(__builtin_amdgcn_s_wait_asynccnt)
    __builtin_amdgcn_s_wait_asynccnt(0);
    asm volatile("" ::: "memory");   // ordering fence for the compiler
#else
    asm volatile("s_wait_asynccnt 0x0" ::: "memory");
#endif
}

// row-major strictly-upper-triangular packed index, matches jnp.triu_indices(27, k=1)
__device__ __forceinline__ int triu_idx(int i, int j) {
    return 26 * i - (i * (i - 1)) / 2 + (j - i - 1);
}

__global__ __launch_bounds__(32 * kWaves)
void dlrm_interact_wmma(const float* __restrict__ dense,
                        const float* __restrict__ sparse,
                        float* __restrict__ out,
                        int B)
{
    __shared__ float smem[kWaves * kRows * kRStride];

    const int tid  = (int)threadIdx.x;
    const int wave = tid >> 5;
    const int lane = tid & 31;
    const int b    = (int)blockIdx.x * kWaves + wave;
    if (b >= B) return;  // wave-uniform: EXEC stays all-1s inside surviving waves

    float* sm = &smem[wave * kRows * kRStride];
    // LDS byte offset of this wave's tile (generic LDS address low 32 bits == DS offset)
    const unsigned smbase = (unsigned)(uintptr_t)(void*)sm;

    // ---- Stage sample tile into LDS with async b128 copies (coalesced) ----
    // Each lane moves 16B: row r floats [lane*4 .. lane*4+3].
    // Row 0: dense feature
    async_copy_b128((const char*)(dense + (size_t)b * kD) + lane * 16,
                    smbase + (unsigned)(0 * kRStride * 4 + lane * 16));
    // Rows 1..26: sparse features (contiguous 26*128 floats per sample)
    const char* sp = (const char*)(sparse + (size_t)b * (kF - 1) * kD) + lane * 16;
    #pragma unroll
    for (int f = 0; f < kF - 1; ++f) {
        async_copy_b128(sp + (size_t)f * (kD * 4),
                        smbase + (unsigned)((1 + f) * kRStride * 4 + lane * 16));
    }
    // Rows 27..31: zero padding (normal DS stores, in-order with wave's DS reads)
    #pragma unroll
    for (int r = kF; r < kRows; ++r) {
        *(float4*)&sm[r * kRStride + lane * 4] = make_float4(0.f, 0.f, 0.f, 0.f);
    }
    // Ensure all async transfers have landed in LDS before fragment reads.
    wait_async_zero();

    // ---- Fragment addressing for V_WMMA_F32_16X16X4_F32 ----
    // A (16x4, MxK): lanes 0-15 hold M=lane, {v0,v1}={K0,K1};
    //               lanes 16-31 hold M=lane-16, {v0,v1}={K2,K3}.
    // B (4x16, KxN) mirrors with N in place of M -> same data for a Gram tile.
    const int n   = lane & 15;
    const int hi  = lane >> 4;      // 0: K0/K1 half, 1: K2/K3 half
    const int klo = hi * 2;

    const float* frag0p = &sm[n * kRStride + klo];          // features 0..15
    const float* frag1p = &sm[(16 + n) * kRStride + klo];   // features 16..31 (27..31 zero)

    v8f acc00 = {};  // rows 0-15  x cols 0-15
    v8f acc01 = {};  // rows 0-15  x cols 16-31
    v8f acc11 = {};  // rows 16-31 x cols 16-31

    #pragma unroll 4
    for (int k = 0; k < kD; k += 4) {
        v2f a0 = *(const v2f*)(frag0p + k);  // ds_load_2addr_b64, conflict-free
        v2f a1 = *(const v2f*)(frag1p + k);
        acc00 = __builtin_amdgcn_wmma_f32_16x16x4_f32(false, a0, false, a0,
                                                      (short)0, acc00, false, false);
        acc01 = __builtin_amdgcn_wmma_f32_16x16x4_f32(false, a0, false, a1,
                                                      (short)0, acc01, false, false);
        acc11 = __builtin_amdgcn_wmma_f32_16x16x4_f32(false, a1, false, a1,
                                                      (short)0, acc11, false, false);
    }

    // ---- Scatter the strictly-upper triangle ----
    // C/D layout: VGPR r -> M = r (lanes 0-15) or M = r+8 (lanes 16-31); N = lane&15.
    float* ob = out + (size_t)b * kPairs;
    #pragma unroll
    for (int r = 0; r < 8; ++r) {
        const int m = r + 8 * hi;
        // tile00: (i, j) = (m, n)
        if (n > m)             ob[triu_idx(m, n)]           = acc00[r];
        // tile01: (i, j) = (m, 16+n); i < j always, need j < 27
        if (n < 11)            ob[triu_idx(m, 16 + n)]      = acc01[r];
        // tile11: (i, j) = (16+m, 16+n); need j < 27 and i < j
        if (n < 11 && n > m)   ob[triu_idx(16 + m, 16 + n)] = acc11[r];
    }
}

extern "C" void kernel_launch(void* const* d_in, const int* in_sizes, int n_in,
                              void* d_out, int out_size, void* d_ws, size_t ws_size,
                              hipStream_t stream) {
    const float* dense  = (const float*)d_in[0];   // (B, 128) f32
    const float* sparse = (const float*)d_in[1];   // (B, 26, 128) f32
    float* out = (float*)d_out;                    // (B, 351) f32

    const int B = in_sizes[0] / kD;
    const int blocks = (B + kWaves - 1) / kWaves;
    hipLaunchKernelGGL(dlrm_interact_wmma, dim3(blocks), dim3(32 * kWaves), 0, stream,
                       dense, sparse, out, B);
}